// MultiHotEmbeddings_12481174962834
// MI455X (gfx1250) — compile-verified
//
#include <hip/hip_runtime.h>
#include <stdint.h>

// MultiHotEmbeddings: 4 embedding-bag(sum) tables, concat -> [4096, 384] f32.
// Bandwidth-bound gather (~186 MB @ 23.3 TB/s => ~8us floor, AI ~ 0.25 F/B):
// no WMMA by design (matrix pipes cannot add HBM bandwidth). CDNA5 paths used:
//  - dim-128 tables: 8-deep b128 gathers w/ SGPR row addressing + global_prefetch_b8
//  - dim-64 tables (L2-resident): double-buffered global_load_async_to_lds_b64
//    pipeline with constant-immediate s_wait_asynccnt overlap.

#define BATCH            4096
#define OUTW             384
#define WAVES_PER_BLOCK  8
#define CHUNK            8    // rows per async chunk (dim-64 path), x2 buffers

__device__ __forceinline__ int rfl(int x) { return __builtin_amdgcn_readfirstlane(x); }

__global__ __launch_bounds__(256) void embbag_fused(
    const int* __restrict__ v0, const int* __restrict__ o0, const float* __restrict__ t0, int nnz0,
    const int* __restrict__ v1, const int* __restrict__ o1, const float* __restrict__ t1, int nnz1,
    const int* __restrict__ v2, const int* __restrict__ o2, const float* __restrict__ t2, int nnz2,
    const int* __restrict__ v3, const int* __restrict__ o3, const float* __restrict__ t3, int nnz3,
    float* __restrict__ out)
{
    // Per-wave private staging: 2 buffers x 8 rows x 64 f32 (256B/row) = 4KB/wave, 32KB/block.
    __shared__ float stage[WAVES_PER_BLOCK][2][CHUNK][64];

    const int wave = threadIdx.x >> 5;
    const int lane = threadIdx.x & 31;
    const int bag  = blockIdx.x * WAVES_PER_BLOCK + wave;

    const int* values; const int* offsets; const float* table;
    int nnz, dim, col0;
    switch (blockIdx.y) {
      case 0:  values = v0; offsets = o0; table = t0; nnz = nnz0; dim = 128; col0 = 0;   break;
      case 1:  values = v1; offsets = o1; table = t1; nnz = nnz1; dim = 128; col0 = 128; break;
      case 2:  values = v2; offsets = o2; table = t2; nnz = nnz2; dim = 64;  col0 = 256; break;
      default: values = v3; offsets = o3; table = t3; nnz = nnz3; dim = 64;  col0 = 320; break;
    }

    // Bag bounds (wave-uniform -> SGPRs). offsets is int32, shape (4096,1).
    const int start = rfl(offsets[bag]);
    const int end   = rfl((bag + 1 < BATCH) ? offsets[bag + 1] : nnz);

    float* outp = out + (size_t)bag * OUTW + col0;

    if (dim == 128) {
        // ---- dim-128 path: lane owns 4 consecutive floats; wave gathers a 512B row per b128.
        // 8 gathers in flight per iteration + next 8 rows prefetched => 8KB/wave pipelined.
        const float* colbase = table + lane * 4;
        float4 acc = make_float4(0.f, 0.f, 0.f, 0.f);

        int i = start;
        for (; i + 8 <= end; i += 8) {
            int r[8];
#pragma unroll
            for (int k = 0; k < 8; ++k) r[k] = rfl(values[i + k]);
            if (i + 16 <= end) {
#pragma unroll
                for (int k = 0; k < 8; ++k) {
                    const int p = rfl(values[i + 8 + k]);
                    __builtin_prefetch(colbase + (size_t)p * 128, 0, 1);
                }
            }
            float4 a[8];
#pragma unroll
            for (int k = 0; k < 8; ++k) a[k] = *(const float4*)(colbase + (size_t)r[k] * 128);
#pragma unroll
            for (int k = 0; k < 8; ++k) {
                acc.x += a[k].x; acc.y += a[k].y; acc.z += a[k].z; acc.w += a[k].w;
            }
        }
        for (; i < end; ++i) {
            const int r = rfl(values[i]);
            const float4 a = *(const float4*)(colbase + (size_t)r * 128);
            acc.x += a.x; acc.y += a.y; acc.z += a.z; acc.w += a.w;
        }
        *(float4*)(outp + lane * 4) = acc;
    } else {
        // ---- dim-64 path: double-buffered async gather-to-LDS pipeline.
        // Wave covers a full 256B row per async op (8B/lane). Completions are in-order on
        // ASYNCcnt, so after issuing CHUNK new ops, s_wait_asynccnt CHUNK <=> prev chunk landed.
        const uint32_t lds_base = (uint32_t)(uintptr_t)(&stage[wave][0][0][0]);
        const uint32_t lane_lds = lds_base + (uint32_t)lane * 8u;
        float2 acc = make_float2(0.f, 0.f);

        int pos = start;
        int cPrev = end - pos; if (cPrev > CHUNK) cPrev = CHUNK;
        int bufPrev = 0;

        if (cPrev > 0) {
            // Prologue: fill buffer 0.
            for (int k = 0; k < cPrev; ++k) {
                const int row = rfl(values[pos + k]);
                const uint64_t ga  = (uint64_t)(const void*)(table + (size_t)row * 64 + lane * 2);
                const uint32_t dst = lane_lds + (uint32_t)k * 256u;
                asm volatile("global_load_async_to_lds_b64 %0, %1, off"
                             :: "v"(dst), "v"(ga) : "memory");
            }
            pos += cPrev;

            while (pos < end) {
                int c = end - pos; if (c > CHUNK) c = CHUNK;
                const int buf = bufPrev ^ 1;
                // Guard: our earlier ds reads of this buffer must retire before the async
                // engine overwrites it.
                asm volatile("s_wait_dscnt 0x0" ::: "memory");
                for (int k = 0; k < c; ++k) {
                    const int row = rfl(values[pos + k]);
                    const uint64_t ga  = (uint64_t)(const void*)(table + (size_t)row * 64 + lane * 2);
                    const uint32_t dst = lane_lds + (uint32_t)(buf * (CHUNK * 256) + k * 256);
                    asm volatile("global_load_async_to_lds_b64 %0, %1, off"
                                 :: "v"(dst), "v"(ga) : "memory");
                }
                pos += c;
                if (c == CHUNK) {
                    asm volatile("s_wait_asynccnt 0x8" ::: "memory");  // prev chunk done
                } else {
                    asm volatile("s_wait_asynccnt 0x0" ::: "memory");  // tail: drain all
                }
                for (int k = 0; k < cPrev; ++k) {
                    const float2 r = *(const float2*)(&stage[wave][bufPrev][k][lane * 2]);
                    acc.x += r.x;
                    acc.y += r.y;
                }
                cPrev = c;
                bufPrev = buf;
            }

            // Epilogue: drain and reduce the last chunk.
            asm volatile("s_wait_asynccnt 0x0" ::: "memory");
            for (int k = 0; k < cPrev; ++k) {
                const float2 r = *(const float2*)(&stage[wave][bufPrev][k][lane * 2]);
                acc.x += r.x;
                acc.y += r.y;
            }
        }
        *(float2*)(outp + lane * 2) = acc;
    }
}

extern "C" void kernel_launch(void* const* d_in, const int* in_sizes, int n_in,
                              void* d_out, int out_size, void* d_ws, size_t ws_size,
                              hipStream_t stream) {
    // setup_inputs() dict order: values0, offsets0, table0, values1, offsets1, table1, ...
    const int*   v0 = (const int*)d_in[0];
    const int*   o0 = (const int*)d_in[1];
    const float* t0 = (const float*)d_in[2];
    const int*   v1 = (const int*)d_in[3];
    const int*   o1 = (const int*)d_in[4];
    const float* t1 = (const float*)d_in[5];
    const int*   v2 = (const int*)d_in[6];
    const int*   o2 = (const int*)d_in[7];
    const float* t2 = (const float*)d_in[8];
    const int*   v3 = (const int*)d_in[9];
    const int*   o3 = (const int*)d_in[10];
    const float* t3 = (const float*)d_in[11];
    float* out = (float*)d_out;

    dim3 grid(BATCH / WAVES_PER_BLOCK, 4);  // 512 blocks x 4 tables; 1 wave = 1 bag
    dim3 block(256);                         // 8 wave32s
    embbag_fused<<<grid, block, 0, stream>>>(
        v0, o0, t0, in_sizes[0],
        v1, o1, t1, in_sizes[3],
        v2, o2, t2, in_sizes[6],
        v3, o3, t3, in_sizes[9],
        out);
}